// TTEmbedding_20761871909369
// MI455X (gfx1250) — compile-verified
//
#include <hip/hip_runtime.h>

typedef float v2f __attribute__((ext_vector_type(2)));
typedef float v8f __attribute__((ext_vector_type(8)));

#define ROWS_PER_BLOCK 256
#define SG_STRIDE 34    // even (8B-aligned v2f loads); 16 consecutive rows hit distinct banks
#define SWP_STRIDE 288  // words per k-pair of packed W; +32 swizzle separates lane halves

// ---------------------------------------------------------------------------
// Kernel 1: collapse core3 (32,4,16) x core4 (16,8,8) x core5 (8,4,1) into
// W[32][128] with out-index o = o0*32 + o1*4 + o2 (matches reference reshapes).
// ---------------------------------------------------------------------------
__global__ void __launch_bounds__(128) tt_build_w(
    const float* __restrict__ core3, const float* __restrict__ core4,
    const float* __restrict__ core5, float* __restrict__ W)
{
    const int o  = threadIdx.x & 127;
    const int o0 = o >> 5, o1 = (o >> 2) & 7, o2 = o & 3;
    float m[16];
#pragma unroll
    for (int a = 0; a < 16; ++a) {
        float acc = 0.f;
#pragma unroll
        for (int b = 0; b < 8; ++b)
            acc = fmaf(core4[a * 64 + o1 * 8 + b], core5[b * 4 + o2], acc);
        m[a] = acc;
    }
#pragma unroll
    for (int r = 0; r < 32; ++r) {
        float acc = 0.f;
#pragma unroll
        for (int a = 0; a < 16; ++a)
            acc = fmaf(core3[r * 64 + o0 * 16 + a], m[a], acc);
        W[r * 128 + o] = acc;
    }
}

// ---------------------------------------------------------------------------
// Kernel 2: per lane, chain the two data-dependent 32x32 matvecs (VALU,
// fully unrolled, float4 gathers that stay hot in L2); stage g in LDS; then
// per wave run the shared-B GEMM g(2x16x32) @ W(32x128) on the matrix pipe
// with V_WMMA_F32_16X16X4_F32; stream results out with NT stores.
//
// W is staged in LDS in a k-pairwise packed layout so each WMMA B fragment
// is a single aligned ds_load_b64 straight into an even register pair.
// ---------------------------------------------------------------------------
__global__ void __launch_bounds__(ROWS_PER_BLOCK) tt_embed(
    const int*   __restrict__ idx,
    const float* __restrict__ core0,
    const float* __restrict__ core1,
    const float* __restrict__ core2,
    const float* __restrict__ W,
    float*       __restrict__ out)
{
    __shared__ float sW[16 * SWP_STRIDE];             // packed W, 18 KB
    __shared__ float sG[ROWS_PER_BLOCK * SG_STRIDE];  // ~34 KB

    const int tid  = threadIdx.x;
    const int lane = tid & 31;
    const int wave = tid >> 5;

    // cache W in LDS, packed: element W[k][n] -> sW[(k>>1)*SWP_STRIDE + n*2 + (k&1)]
#pragma unroll
    for (int i = 0; i < 16; ++i) {
        const int f = tid + i * ROWS_PER_BLOCK;       // flat index into W (32x128)
        const int k = f >> 7, n = f & 127;
        sW[(k >> 1) * SWP_STRIDE + n * 2 + (k & 1)] = W[f];
    }

    const int rowblk = blockIdx.x * ROWS_PER_BLOCK;
    const int rowloc = wave * 32 + lane;

    // ---- per-lane TT contraction chain ------------------------------------
    const int id  = idx[rowblk + rowloc];
    const int i0  = id / 10000;
    const int rem = id - i0 * 10000;
    const int i1  = rem / 100;
    const int i2  = rem - i1 * 100;

    float ga[32], gb[32];
    {
        const float4* p = (const float4*)(core0 + i0 * 32);
#pragma unroll
        for (int q = 0; q < 8; ++q) {
            float4 v = p[q];
            ga[q*4+0] = v.x; ga[q*4+1] = v.y; ga[q*4+2] = v.z; ga[q*4+3] = v.w;
        }
    }
#pragma unroll
    for (int s = 0; s < 32; ++s) gb[s] = 0.f;
    {
        const float* base = core1 + i1 * 32;          // core1[r, i1, s] = r*3200 + i1*32 + s
#pragma unroll
        for (int r = 0; r < 32; ++r) {
            const float4* p = (const float4*)(base + r * 3200);
            const float gr = ga[r];
#pragma unroll
            for (int q = 0; q < 8; ++q) {
                float4 v = p[q];
                gb[q*4+0] = fmaf(gr, v.x, gb[q*4+0]);
                gb[q*4+1] = fmaf(gr, v.y, gb[q*4+1]);
                gb[q*4+2] = fmaf(gr, v.z, gb[q*4+2]);
                gb[q*4+3] = fmaf(gr, v.w, gb[q*4+3]);
            }
        }
    }
#pragma unroll
    for (int s = 0; s < 32; ++s) ga[s] = 0.f;
    {
        const float* base = core2 + i2 * 32;
#pragma unroll
        for (int r = 0; r < 32; ++r) {
            const float4* p = (const float4*)(base + r * 3200);
            const float gr = gb[r];
#pragma unroll
            for (int q = 0; q < 8; ++q) {
                float4 v = p[q];
                ga[q*4+0] = fmaf(gr, v.x, ga[q*4+0]);
                ga[q*4+1] = fmaf(gr, v.y, ga[q*4+1]);
                ga[q*4+2] = fmaf(gr, v.z, ga[q*4+2]);
                ga[q*4+3] = fmaf(gr, v.w, ga[q*4+3]);
            }
        }
    }

    // stage final g row into LDS
    float* grow = sG + rowloc * SG_STRIDE;
#pragma unroll
    for (int s = 0; s < 32; ++s) grow[s] = ga[s];

    __syncthreads();

    // ---- WMMA GEMM: two 16-row tiles per wave -----------------------------
    // A 16x4 f32 layout: lanes 0-15 hold M=lane, K=4j..4j+1; lanes 16-31 K=4j+2..3.
    // B  4x16 f32 layout mirrored (N = lane%16). C: VGPR v -> M=v (+8 upper half).
    const int nlo   = lane & 15;
    const int khalf = lane >> 4;     // 0 | 1
    const int mh    = khalf * 8;
    const float* a0base = sG + (wave * 32 + nlo) * SG_STRIDE;
    const float* a1base = a0base + 16 * SG_STRIDE;
    float* out0 = out + (size_t)(rowblk + wave * 32) * 128;

#pragma unroll
    for (int c = 0; c < 8; ++c) {
        v8f acc0 = {0.f,0.f,0.f,0.f,0.f,0.f,0.f,0.f};
        v8f acc1 = {0.f,0.f,0.f,0.f,0.f,0.f,0.f,0.f};
#pragma unroll
        for (int j = 0; j < 8; ++j) {
            const int kb = j * 4 + khalf * 2;                 // even
            v2f a0 = *(const v2f*)(a0base + kb);              // ds_load_b64
            v2f a1 = *(const v2f*)(a1base + kb);              // ds_load_b64
            v2f b  = *(const v2f*)(sW + (2 * j + khalf) * SWP_STRIDE
                                      + (c * 16 + nlo) * 2);  // ds_load_b64
            acc0 = __builtin_amdgcn_wmma_f32_16x16x4_f32(false, a0, false, b,
                                                         (short)0, acc0, false, false);
            acc1 = __builtin_amdgcn_wmma_f32_16x16x4_f32(false, a1, false, b,
                                                         (short)0, acc1, false, false);
        }
#pragma unroll
        for (int v = 0; v < 8; ++v) {
            __builtin_nontemporal_store(acc0[v], out0 + (mh + v) * 128 + c * 16 + nlo);
            __builtin_nontemporal_store(acc1[v], out0 + (16 + mh + v) * 128 + c * 16 + nlo);
        }
    }
}

// ---------------------------------------------------------------------------
extern "C" void kernel_launch(void* const* d_in, const int* in_sizes, int n_in,
                              void* d_out, int out_size, void* d_ws, size_t ws_size,
                              hipStream_t stream) {
    const int*   idx   = (const int*)  d_in[0];
    const float* core0 = (const float*)d_in[1];
    const float* core1 = (const float*)d_in[2];
    const float* core2 = (const float*)d_in[3];
    const float* core3 = (const float*)d_in[4];
    const float* core4 = (const float*)d_in[5];
    const float* core5 = (const float*)d_in[6];
    float* W   = (float*)d_ws;          // 32*128 f32 = 16 KB scratch
    float* out = (float*)d_out;

    const int nrows = in_sizes[0];      // 512*128 = 65536, divisible by 256

    tt_build_w<<<1, 128, 0, stream>>>(core3, core4, core5, W);
    tt_embed<<<nrows / ROWS_PER_BLOCK, ROWS_PER_BLOCK, 0, stream>>>(
        idx, core0, core1, core2, W, out);
}